// PropertyPredictor_54992761258163
// MI455X (gfx1250) — compile-verified
//
#include <hip/hip_runtime.h>
#include <hip/hip_bf16.h>

typedef __attribute__((ext_vector_type(2))) float v2f;
typedef __attribute__((ext_vector_type(8))) float v8f;

#define N_NODES_C   524288
#define N_EDGES_C   786432
#define NUM_GRAPHS_C 8192
#define NTYPES      16
#define NODE_DIM_C  256
#define EDGE_DIM_C  128
#define HDIM        384   // NODE_DIM + EDGE_DIM
#define KDIM        48    // 16 node feats + 32 edge feats (pre-embedding, folded)

__device__ __forceinline__ int lower_bound_i(const int* __restrict__ a, int n, int key) {
  int lo = 0, hi = n;
  while (lo < hi) {
    int mid = (lo + hi) >> 1;
    if (a[mid] < key) lo = mid + 1; else hi = mid;
  }
  return lo;
}

// Wc[k][n] = (k<16) ? sum_j W_node[k][j]*W1[j][n] : sum_j W_edge[k-16][j]*W1[256+j][n]
__global__ void fold_weights_kernel(const float* __restrict__ W_node,
                                    const float* __restrict__ W_edge,
                                    const float* __restrict__ W1,
                                    float* __restrict__ Wc) {
  int k = blockIdx.x;       // 0..47
  int n = threadIdx.x;      // 0..383
  float acc = 0.f;
  if (k < NTYPES) {
    const float* wrow = W_node + k * NODE_DIM_C;
    #pragma unroll 4
    for (int j = 0; j < NODE_DIM_C; ++j) acc += wrow[j] * W1[j * HDIM + n];
  } else {
    const float* wrow = W_edge + (k - NTYPES) * EDGE_DIM_C;
    #pragma unroll 4
    for (int j = 0; j < EDGE_DIM_C; ++j) acc += wrow[j] * W1[(NODE_DIM_C + j) * HDIM + n];
  }
  Wc[k * HDIM + n] = acc;
}

// Per-graph mean of raw node features (batch_node is sorted -> binary search, no atomics).
__global__ void node_pool_kernel(const float* __restrict__ h_node,
                                 const int* __restrict__ batch_node,
                                 float* __restrict__ P) {
  __shared__ float sAcc[128];
  const int g = blockIdx.x;
  const int tid = threadIdx.x;          // 128 threads
  const int f = tid & 15, chunk = tid >> 4;   // 16 feats x 8 row-chunks
  const int lo = lower_bound_i(batch_node, N_NODES_C, g);
  const int hi = lower_bound_i(batch_node, N_NODES_C, g + 1);
  float acc = 0.f;
  for (int i = lo + chunk; i < hi; i += 8) acc += h_node[i * 16 + f];
  sAcc[tid] = acc;
  __syncthreads();
  if (tid < 16) {
    float s = 0.f;
    #pragma unroll
    for (int c = 0; c < 8; ++c) s += sAcc[c * 16 + tid];
    int cnt = hi - lo;
    float inv = (cnt > 0) ? 1.0f / (float)cnt : 0.0f;   // matches max(cnt,1) (sum==0 when empty)
    P[g * KDIM + tid] = s * inv;
  }
}

// Per-graph mean of concat(h_node[src], h_node[dst]) (batch_edge sorted).
__global__ void edge_pool_kernel(const float* __restrict__ h_node,
                                 const int* __restrict__ edge_index,
                                 const int* __restrict__ batch_edge,
                                 float* __restrict__ P) {
  __shared__ float sAcc[128];
  const int g = blockIdx.x;
  const int tid = threadIdx.x;          // 128 threads
  const int f = tid & 31, chunk = tid >> 5;   // 32 feats x 4 edge-chunks
  const int lo = lower_bound_i(batch_edge, N_EDGES_C, g);
  const int hi = lower_bound_i(batch_edge, N_EDGES_C, g + 1);
  float acc = 0.f;
  for (int e = lo + chunk; e < hi; e += 4) {
    int src = edge_index[e];
    int dst = edge_index[N_EDGES_C + e];
    int node = (f < 16) ? src : dst;          // h_node is L2-resident (33.5 MB)
    acc += h_node[node * 16 + (f & 15)];
  }
  sAcc[tid] = acc;
  __syncthreads();
  if (tid < 32) {
    float s = sAcc[tid] + sAcc[32 + tid] + sAcc[64 + tid] + sAcc[96 + tid];
    int cnt = hi - lo;
    float inv = (cnt > 0) ? 1.0f / (float)cnt : 0.0f;
    P[g * KDIM + 16 + tid] = s * inv;
  }
}

// pred[m] = relu(P[m,:48] @ Wc + b1) @ W2 + b2, one 16-row tile per block.
// 24 waves per block, wave w owns N-tile w (cols 16w..16w+15). f32 WMMA, K=48.
__global__ void mlp_wmma_kernel(const float* __restrict__ P,
                                const float* __restrict__ Wc,
                                const float* __restrict__ b1,
                                const float* __restrict__ W2,
                                const float* __restrict__ b2,
                                float* __restrict__ pred) {
  __shared__ float sP[16][KDIM + 1];   // +1 pad: 49-stride dodges 64-bank conflicts
  __shared__ float sPred[24][16];
  const int tid = threadIdx.x;         // 768 = 16*48 = 24 waves
  const int m0 = blockIdx.x * 16;

  { // cooperative load of the 16x48 activation tile
    int row = tid / KDIM, col = tid % KDIM;
    sP[row][col] = P[(m0 + row) * KDIM + col];
  }
  __syncthreads();

  const int w    = tid >> 5;     // wave id == N-tile id (0..23)
  const int l    = tid & 31;
  const int half = l >> 4;       // K-half select per ISA layout
  const int lrow = l & 15;       // A row (lanes 0-15 / 16-31 both cover M=0..15)
  const int n    = w * 16 + lrow; // B/C/D column for this lane

  v8f c = {};
  #pragma unroll
  for (int kk = 0; kk < KDIM; kk += 4) {
    const int kb = kk + 2 * half;        // K = 2*half + vgpr  (16x4 f32 A layout)
    v2f a, b;
    a.x = sP[lrow][kb];
    a.y = sP[lrow][kb + 1];
    b.x = Wc[kb * HDIM + n];             // B 4x16: row K striped across lanes
    b.y = Wc[(kb + 1) * HDIM + n];
    c = __builtin_amdgcn_wmma_f32_16x16x4_f32(
        /*neg_a=*/false, a, /*neg_b=*/false, b,
        /*c_mod=*/(short)0, c, /*reuse_a=*/false, /*reuse_b=*/false);
  }

  // C/D layout: VGPR r, lane-half h -> row m = r + 8h, col n = lane&15 (+ tile offset).
  const float bias = b1[n];
  const float w2v  = W2[n];
  float pd[8];
  #pragma unroll
  for (int r = 0; r < 8; ++r) {
    float z = c[r] + bias;
    z = z > 0.f ? z : 0.f;
    pd[r] = z * w2v;                      // partial of the W2 dot for row r+8*half
  }
  // reduce across the 16 lanes of each half (xor masks < 16 stay within a half)
  #pragma unroll
  for (int r = 0; r < 8; ++r) {
    float v = pd[r];
    v += __shfl_xor(v, 1, 32);
    v += __shfl_xor(v, 2, 32);
    v += __shfl_xor(v, 4, 32);
    v += __shfl_xor(v, 8, 32);
    pd[r] = v;
  }
  if (lrow == 0) {
    #pragma unroll
    for (int r = 0; r < 8; ++r) sPred[w][half * 8 + r] = pd[r];
  }
  __syncthreads();

  if (tid < 16) {
    float s = b2[0];
    #pragma unroll
    for (int ww = 0; ww < 24; ++ww) s += sPred[ww][tid];
    pred[m0 + tid] = s;
  }
}

extern "C" void kernel_launch(void* const* d_in, const int* in_sizes, int n_in,
                              void* d_out, int out_size, void* d_ws, size_t ws_size,
                              hipStream_t stream) {
  const float* h_node     = (const float*)d_in[0];
  /* pos_node d_in[1] is unused by the reference */
  const int*   batch_node = (const int*)d_in[2];
  const int*   edge_index = (const int*)d_in[3];
  const int*   batch_edge = (const int*)d_in[4];
  const float* W_node     = (const float*)d_in[5];
  const float* W_edge     = (const float*)d_in[6];
  const float* W1         = (const float*)d_in[7];
  const float* b1         = (const float*)d_in[8];
  const float* W2         = (const float*)d_in[9];
  const float* b2         = (const float*)d_in[10];
  float* pred = (float*)d_out;

  float* P  = (float*)d_ws;                              // [8192 * 48]
  float* Wc = P + (size_t)NUM_GRAPHS_C * KDIM;           // [48 * 384]

  hipLaunchKernelGGL(fold_weights_kernel, dim3(KDIM), dim3(HDIM), 0, stream,
                     W_node, W_edge, W1, Wc);
  hipLaunchKernelGGL(node_pool_kernel, dim3(NUM_GRAPHS_C), dim3(128), 0, stream,
                     h_node, batch_node, P);
  hipLaunchKernelGGL(edge_pool_kernel, dim3(NUM_GRAPHS_C), dim3(128), 0, stream,
                     h_node, edge_index, batch_edge, P);
  hipLaunchKernelGGL(mlp_wmma_kernel, dim3(NUM_GRAPHS_C / 16), dim3(768), 0, stream,
                     P, Wc, b1, W2, b2, pred);
}